// QuantumAttentionHead_60687887892883
// MI455X (gfx1250) — compile-verified
//
#include <hip/hip_runtime.h>
#include <math.h>

// ---------------------------------------------------------------------------
// QuantumAttentionHead for MI455X (gfx1250, wave32).
//   q = x@Wq^T+bq ; k = x@Wk^T+bk ; v = x@Wv^T+bv         (WMMA f32 16x16x4)
//   q_red = q@Wqq^T ; k_red = k@Wkq^T                      (tiny VALU)
//   scores[b,i,j] = Ws . expval( 4-qubit circuit(tanh(qr_i*kr_j)) ) + bs
//   attn = softmax(scores) ; ctx = attn@v ; out = ctx@Wo^T+bo (WMMA f32)
// All f32 to match the reference; the quantum sim dominates (pure VALU),
// so f32 WMMA is the right precision/perf tradeoff for the GEMMs.
// GEMM uses 2x2 register tiling per wave: 4 loads -> 4 WMMAs per K-step.
// ---------------------------------------------------------------------------

#define B_  2
#define S_  512
#define E_  512
#define Q_  64
#define NQ_ 4

typedef __attribute__((ext_vector_type(2))) float v2f;
typedef __attribute__((ext_vector_type(8))) float v8f;

// ---------------------------------------------------------------------------
// Generic WMMA f32 GEMM:  C[m][n] = sum_k A[m][k]*B(k,n) + bias[n]
//   BT=true : B stored as weight W[n][k] (compute A @ W^T)
//   BT=false: B stored row-major B[k][n] (compute A @ B)
// Wave -> 32x32 tile (2x2 of 16x16 WMMA tiles); block = 4 waves (2x2) -> 64x64.
// Fragment layouts per CDNA5 ISA 7.12.2 (f32 16x4 A, 4x16 B, 16x16 C/D).
// ---------------------------------------------------------------------------
template<bool BT>
__global__ void __launch_bounds__(128)
gemm_wmma_f32(const float* __restrict__ A, const float* __restrict__ Bm,
              const float* __restrict__ bias, float* __restrict__ C,
              int M, int N, int K, long sAb, long sBb, long sCb)
{
    const int lane = threadIdx.x & 31;
    const int wave = threadIdx.x >> 5;
    const int half = lane >> 4;        // 0: lanes 0-15, 1: lanes 16-31
    const int l16  = lane & 15;
    const int m0   = (blockIdx.y << 6) + ((wave >> 1) << 5);   // wave row -> 32 rows
    const int n0   = (blockIdx.x << 6) + ((wave & 1) << 5);    // wave col -> 32 cols

    A  += (long)blockIdx.z * sAb;
    Bm += (long)blockIdx.z * sBb;
    C  += (long)blockIdx.z * sCb;

    const int ar0 = m0 + l16;          // A rows held by this lane (two M tiles)
    const int ar1 = m0 + 16 + l16;
    const int nc0 = n0 + l16;          // B/C cols held by this lane (two N tiles)
    const int nc1 = n0 + 16 + l16;

    v8f acc00 = {}, acc01 = {}, acc10 = {}, acc11 = {};
    for (int kb = 0; kb < K; kb += 4) {
        const int ka = kb + (half << 1);       // lanes<16 -> K=kb,kb+1 ; else kb+2,kb+3
        v2f a0 = *(const v2f*)(A + (long)ar0 * K + ka);
        v2f a1 = *(const v2f*)(A + (long)ar1 * K + ka);
        v2f b0, b1;
        if (BT) {
            b0 = *(const v2f*)(Bm + (long)nc0 * K + ka);
            b1 = *(const v2f*)(Bm + (long)nc1 * K + ka);
        } else {
            b0.x = Bm[(long)(ka    ) * N + nc0];
            b0.y = Bm[(long)(ka + 1) * N + nc0];
            b1.x = Bm[(long)(ka    ) * N + nc1];
            b1.y = Bm[(long)(ka + 1) * N + nc1];
        }
        acc00 = __builtin_amdgcn_wmma_f32_16x16x4_f32(false, a0, false, b0, (short)0, acc00, false, false);
        acc01 = __builtin_amdgcn_wmma_f32_16x16x4_f32(false, a0, false, b1, (short)0, acc01, false, false);
        acc10 = __builtin_amdgcn_wmma_f32_16x16x4_f32(false, a1, false, b0, (short)0, acc10, false, false);
        acc11 = __builtin_amdgcn_wmma_f32_16x16x4_f32(false, a1, false, b1, (short)0, acc11, false, false);
    }

    const float bv0 = bias ? bias[nc0] : 0.0f;
    const float bv1 = bias ? bias[nc1] : 0.0f;
#pragma unroll
    for (int r = 0; r < 8; ++r) {
        const int mlo = r + (half << 3);       // VGPR r: M=r (lo half) / r+8 (hi half)
        C[(long)(m0      + mlo) * N + nc0] = acc00[r] + bv0;
        C[(long)(m0      + mlo) * N + nc1] = acc01[r] + bv1;
        C[(long)(m0 + 16 + mlo) * N + nc0] = acc10[r] + bv0;
        C[(long)(m0 + 16 + mlo) * N + nc1] = acc11[r] + bv1;
    }
}

// ---------------------------------------------------------------------------
// q_red / k_red: (1024,64) @ (4,64)^T + bias -> (1024,4). One thread per row.
// ---------------------------------------------------------------------------
__global__ void reduce_qk_kernel(const float* __restrict__ q, const float* __restrict__ k,
                                 const float* __restrict__ Wqq, const float* __restrict__ bqq,
                                 const float* __restrict__ Wkq, const float* __restrict__ bkq,
                                 float* __restrict__ qr, float* __restrict__ kr)
{
    const int row = blockIdx.x * blockDim.x + threadIdx.x;
    if (row >= B_ * S_) return;
#pragma unroll
    for (int nq = 0; nq < NQ_; ++nq) {
        float aq = bqq[nq], ak = bkq[nq];
        for (int t = 0; t < Q_; ++t) {
            aq = fmaf(q[row * Q_ + t], Wqq[nq * Q_ + t], aq);
            ak = fmaf(k[row * Q_ + t], Wkq[nq * Q_ + t], ak);
        }
        qr[row * NQ_ + nq] = aq;
        kr[row * NQ_ + nq] = ak;
    }
}

// ---------------------------------------------------------------------------
// 4-qubit state-vector gates. State = 16 complex amplitudes in registers.
// Flat index bit (3-wire) selects the wire's axis (state[N,2,2,2,2]).
// All loops unrolled so every index constant-folds -> state stays in VGPRs.
// ---------------------------------------------------------------------------
__device__ __forceinline__ void gate_ry(float* re, float* im, const int wire, float half) {
    float s, c; __sincosf(half, &s, &c);
    const int st = 1 << (3 - wire);
#pragma unroll
    for (int p = 0; p < 8; ++p) {
        const int low = p & (st - 1);
        const int i0  = ((p - low) << 1) | low;    // insert 0 at bit (3-wire)
        const int i1  = i0 | st;
        const float ar = re[i0], ai = im[i0], br = re[i1], bi = im[i1];
        re[i0] = c * ar - s * br;  im[i0] = c * ai - s * bi;
        re[i1] = s * ar + c * br;  im[i1] = s * ai + c * bi;
    }
}

__device__ __forceinline__ void gate_rx(float* re, float* im, const int wire, float half) {
    float s, c; __sincosf(half, &s, &c);
    const int st = 1 << (3 - wire);
#pragma unroll
    for (int p = 0; p < 8; ++p) {
        const int low = p & (st - 1);
        const int i0  = ((p - low) << 1) | low;
        const int i1  = i0 | st;
        const float ar = re[i0], ai = im[i0], br = re[i1], bi = im[i1];
        // [[c, -i s],[-i s, c]]
        re[i0] = c * ar + s * bi;  im[i0] = c * ai - s * br;
        re[i1] = c * br + s * ai;  im[i1] = c * bi - s * ar;
    }
}

__device__ __forceinline__ void gate_rz(float* re, float* im, const int wire, float half) {
    float s, c; __sincosf(half, &s, &c);
    const int st = 1 << (3 - wire);
#pragma unroll
    for (int p = 0; p < 16; ++p) {
        const float r = re[p], m = im[p];
        if (p & st) { re[p] = c * r - s * m;  im[p] = c * m + s * r; }   // e^{+i h}
        else        { re[p] = c * r + s * m;  im[p] = c * m - s * r; }   // e^{-i h}
    }
}

__device__ __forceinline__ void gate_cnot(float* re, float* im, const int cw, const int tw) {
    const int cm = 1 << (3 - cw), tm = 1 << (3 - tw);
#pragma unroll
    for (int p = 0; p < 16; ++p) {
        if ((p & cm) && !(p & tm)) {
            const int q = p | tm;
            const float tr = re[p], ti = im[p];
            re[p] = re[q]; im[p] = im[q];
            re[q] = tr;    im[q] = ti;
        }
    }
}

// ---------------------------------------------------------------------------
// Quantum scores: one thread per (b,i,j).
//   in = tanh(qr[b,i,:] * kr[b,j,:]); circuit; score = Ws.expv + bs
// ---------------------------------------------------------------------------
__global__ void __launch_bounds__(256)
quantum_scores_kernel(const float* __restrict__ qr, const float* __restrict__ kr,
                      const float* __restrict__ w_rx, const float* __restrict__ w_ry,
                      const float* __restrict__ w_rz, const float* __restrict__ w_fin,
                      const float* __restrict__ Ws,   const float* __restrict__ bs,
                      float* __restrict__ scores)
{
    const int idx = blockIdx.x * blockDim.x + threadIdx.x;
    const int total = B_ * S_ * S_;
    if (idx >= total) return;
    const int b   = idx / (S_ * S_);
    const int rem = idx - b * S_ * S_;
    const int i   = rem / S_;
    const int j   = rem - i * S_;

    const float4 qv = *(const float4*)(qr + (long)(b * S_ + i) * NQ_);
    const float4 kv = *(const float4*)(kr + (long)(b * S_ + j) * NQ_);
    float in[NQ_];
    in[0] = tanhf(qv.x * kv.x);  in[1] = tanhf(qv.y * kv.y);
    in[2] = tanhf(qv.z * kv.z);  in[3] = tanhf(qv.w * kv.w);

    float re[16], im[16];
#pragma unroll
    for (int p = 0; p < 16; ++p) { re[p] = 0.0f; im[p] = 0.0f; }
    re[0] = 1.0f;

    const float HPI = 1.57079632679489662f;  // pi/2
#pragma unroll
    for (int q = 0; q < NQ_; ++q) gate_ry(re, im, q, in[q] * HPI);

#pragma unroll
    for (int l = 0; l < 2; ++l) {
#pragma unroll
        for (int q = 0; q < NQ_; ++q) {
            gate_rx(re, im, q, 0.5f * w_rx[l * NQ_ + q]);
            gate_ry(re, im, q, 0.5f * w_ry[l * NQ_ + q]);
            gate_rz(re, im, q, 0.5f * w_rz[l * NQ_ + q]);
        }
        gate_cnot(re, im, 0, 1);
        gate_cnot(re, im, 1, 2);
        gate_cnot(re, im, 2, 3);
        gate_cnot(re, im, 3, 0);
    }
#pragma unroll
    for (int q = 0; q < NQ_; ++q) gate_ry(re, im, q, 0.5f * w_fin[q]);

    float ez[NQ_] = {0.f, 0.f, 0.f, 0.f};
#pragma unroll
    for (int p = 0; p < 16; ++p) {
        const float pr = re[p] * re[p] + im[p] * im[p];
#pragma unroll
        for (int q = 0; q < NQ_; ++q)
            ez[q] += (p & (1 << (3 - q))) ? -pr : pr;
    }

    scores[idx] = ez[0] * Ws[0] + ez[1] * Ws[1] + ez[2] * Ws[2] + ez[3] * Ws[3] + bs[0];
}

// ---------------------------------------------------------------------------
// In-place row softmax over last axis: (B*S) rows of length S.
// ---------------------------------------------------------------------------
__global__ void __launch_bounds__(256)
softmax_kernel(float* __restrict__ scores)
{
    float* p = scores + (long)blockIdx.x * S_;
    __shared__ float red[256];
    const int t = threadIdx.x;

    float m = -3.402823466e38f;
    for (int c = t; c < S_; c += 256) m = fmaxf(m, p[c]);
    red[t] = m; __syncthreads();
    for (int o = 128; o > 0; o >>= 1) {
        if (t < o) red[t] = fmaxf(red[t], red[t + o]);
        __syncthreads();
    }
    m = red[0]; __syncthreads();

    float sum = 0.0f;
    for (int c = t; c < S_; c += 256) {
        const float e = __expf(p[c] - m);
        p[c] = e;
        sum += e;
    }
    red[t] = sum; __syncthreads();
    for (int o = 128; o > 0; o >>= 1) {
        if (t < o) red[t] += red[t + o];
        __syncthreads();
    }
    const float inv = 1.0f / red[0];
    for (int c = t; c < S_; c += 256) p[c] *= inv;
}

// ---------------------------------------------------------------------------
extern "C" void kernel_launch(void* const* d_in, const int* in_sizes, int n_in,
                              void* d_out, int out_size, void* d_ws, size_t ws_size,
                              hipStream_t stream)
{
    const float* x    = (const float*)d_in[0];
    const float* Wq   = (const float*)d_in[1];
    const float* bq   = (const float*)d_in[2];
    const float* Wk   = (const float*)d_in[3];
    const float* bk   = (const float*)d_in[4];
    const float* Wv   = (const float*)d_in[5];
    const float* bv   = (const float*)d_in[6];
    const float* Wqq  = (const float*)d_in[7];
    const float* bqq  = (const float*)d_in[8];
    const float* Wkq  = (const float*)d_in[9];
    const float* bkq  = (const float*)d_in[10];
    const float* w_rx = (const float*)d_in[11];
    const float* w_ry = (const float*)d_in[12];
    const float* w_rz = (const float*)d_in[13];
    const float* w_fin= (const float*)d_in[14];
    const float* Ws   = (const float*)d_in[15];
    const float* bs   = (const float*)d_in[16];
    const float* Wo   = (const float*)d_in[17];
    const float* bo   = (const float*)d_in[18];
    float* out = (float*)d_out;

    // Workspace carve-up (floats): ~6.9 MB total.
    float* ws   = (float*)d_ws;
    float* q    = ws;                         // 1024*64
    float* k    = q    + (B_*S_) * Q_;        // 1024*64
    float* qr   = k    + (B_*S_) * Q_;        // 1024*4
    float* kr   = qr   + (B_*S_) * NQ_;       // 1024*4
    float* v    = kr   + (B_*S_) * NQ_;       // 1024*512
    float* sc   = v    + (B_*S_) * E_;        // 2*512*512 (scores -> attn in place)
    float* ctx  = sc   + (long)B_*S_*S_;      // 1024*512

    const int M = B_ * S_;                    // 1024

    // q = x@Wq^T+bq, k = x@Wk^T+bk  (N=64), v = x@Wv^T+bv (N=512)
    gemm_wmma_f32<true><<<dim3(Q_/64, M/64, 1), 128, 0, stream>>>(
        x, Wq, bq, q, M, Q_, E_, 0, 0, 0);
    gemm_wmma_f32<true><<<dim3(Q_/64, M/64, 1), 128, 0, stream>>>(
        x, Wk, bk, k, M, Q_, E_, 0, 0, 0);
    gemm_wmma_f32<true><<<dim3(E_/64, M/64, 1), 128, 0, stream>>>(
        x, Wv, bv, v, M, E_, E_, 0, 0, 0);

    // q_red / k_red
    reduce_qk_kernel<<<dim3((M + 255) / 256), 256, 0, stream>>>(
        q, k, Wqq, bqq, Wkq, bkq, qr, kr);

    // quantum circuit per (b,i,j) -> scores
    const int total = B_ * S_ * S_;
    quantum_scores_kernel<<<dim3((total + 255) / 256), 256, 0, stream>>>(
        qr, kr, w_rx, w_ry, w_rz, w_fin, Ws, bs, sc);

    // softmax rows
    softmax_kernel<<<dim3(B_ * S_), 256, 0, stream>>>(sc);

    // ctx = attn @ v   (batched over B, B[k][n] row-major)
    gemm_wmma_f32<false><<<dim3(E_/64, S_/64, B_), 128, 0, stream>>>(
        sc, v, nullptr, ctx, S_, E_, S_,
        (long)S_ * S_, (long)S_ * E_, (long)S_ * E_);

    // out = ctx @ Wo^T + bo
    gemm_wmma_f32<true><<<dim3(E_/64, M/64, 1), 128, 0, stream>>>(
        ctx, Wo, bo, out, M, E_, E_, 0, 0, 0);
}